// Model_39676907881797
// MI455X (gfx1250) — compile-verified
//
#include <hip/hip_runtime.h>

// CDNA5 / gfx1250, wave32. V_WMMA_F32_16X16X4_F32 for the attn@v product.
// Key trick this round: JAX threefry2x32 counter pair (f, f+H) yields the
// dropout bits for BOTH (i, j) and (i+4096, j); processing the row pair
// together halves the PRNG work, which is the dominant cost.
typedef __attribute__((ext_vector_type(2))) float v2f;
typedef __attribute__((ext_vector_type(8))) float v8f;

#define NCOL   8192
#define ROWS_PER_BLOCK 16              // per row-group; block owns 2 groups
#define STILE  2048
#define LOG2E  1.4426950408889634f

__device__ __forceinline__ unsigned tf_rotl(unsigned x, unsigned r) {
  return (x << r) | (x >> (32u - r));
}

// Bit-exact JAX threefry2x32-20 for jax.random.key(42) -> key = (0, 42).
// f = i*8192 + j with i < 4096 (so f < H). Counter pair = (f, f+H).
// Returns {x0, x1} = uniform bits for rows i and i+4096 at column j.
__device__ __forceinline__ uint2 jax_threefry_pair(unsigned f) {
  const unsigned H  = 33554432u;                 // 8192*8192/2
  const unsigned k0 = 0u, k1 = 42u;
  const unsigned k2 = 0x1BD11BDAu ^ k0 ^ k1;
  unsigned x0 = f + k0;
  unsigned x1 = (f + H) + k1;
#define TF_R4(a,b,c,d)                                     \
  x0 += x1; x1 = tf_rotl(x1,a); x1 ^= x0;                  \
  x0 += x1; x1 = tf_rotl(x1,b); x1 ^= x0;                  \
  x0 += x1; x1 = tf_rotl(x1,c); x1 ^= x0;                  \
  x0 += x1; x1 = tf_rotl(x1,d); x1 ^= x0;
  TF_R4(13,15,26, 6)  x0 += k1; x1 += k2 + 1u;
  TF_R4(17,29,16,24)  x0 += k2; x1 += k0 + 2u;
  TF_R4(13,15,26, 6)  x0 += k0; x1 += k1 + 3u;
  TF_R4(17,29,16,24)  x0 += k1; x1 += k2 + 4u;
  TF_R4(13,15,26, 6)  x0 += k2; x1 += k0 + 5u;
#undef TF_R4
  return make_uint2(x0, x1);
}

__device__ __forceinline__ bool keep_bits(unsigned bits) {
  // JAX uniform: bitcast((bits>>9)|0x3f800000) - 1.0; keep if u < 0.9
  float u = __uint_as_float((bits >> 9) | 0x3f800000u) - 1.0f;
  return u < 0.9f;
}

__global__ __launch_bounds__(256)
void attn_dim1_kernel(const float* __restrict__ x1, const float* __restrict__ x2,
                      const float* __restrict__ wq, const float* __restrict__ wk,
                      const float* __restrict__ wv, float* __restrict__ out) {
  __shared__ float x1s[STILE];          // staged column slab of x1 (for v)
  __shared__ float x2s[STILE];          // staged column slab of x2 (for k)
  __shared__ float accS[2 * 256 * 8];   // two D-tile sets (row groups A/B)
  __shared__ float zS[2 * 256];         // two denominator sets
  __shared__ float s_mx, s_mn;

  const int tid  = threadIdx.x;
  const int w    = tid >> 5;            // wave id (wave32)
  const int lane = tid & 31;
  const int r    = lane & 15;           // A-row / B-col index
  const int kk   = (lane >> 4) << 1;    // K slots {kk, kk+1} per 16x16x4 layout

  // ---- global max/min of x2 (exact softmax row-max for rank-1 scores) ----
  float mx = -3.402823466e38f, mn = 3.402823466e38f;
  for (int j = tid; j < NCOL; j += 256) {
    float v = x2[j]; mx = fmaxf(mx, v); mn = fminf(mn, v);
  }
  zS[tid] = mx; __syncthreads();
  for (int s = 128; s > 0; s >>= 1) {
    if (tid < s) zS[tid] = fmaxf(zS[tid], zS[tid + s]);
    __syncthreads();
  }
  if (tid == 0) s_mx = zS[0];
  __syncthreads();
  zS[tid] = mn; __syncthreads();
  for (int s = 128; s > 0; s >>= 1) {
    if (tid < s) zS[tid] = fminf(zS[tid], zS[tid + s]);
    __syncthreads();
  }
  if (tid == 0) s_mn = zS[0];
  __syncthreads();

  float c = 0.f;                        // dot(wq, wk) -- uniform scalar loads
  for (int h = 0; h < 8; ++h) c += wq[h] * wk[h];

  const unsigned iA = blockIdx.x * ROWS_PER_BLOCK + r;    // 0..4095
  const unsigned iB = iA + 4096u;                         // paired row
  const float biA = c * x1[iA] * LOG2E;
  const float biB = c * x1[iB] * LOG2E;
  const float rmA = fmaxf(biA * s_mx, biA * s_mn);
  const float rmB = fmaxf(biB * s_mx, biB * s_mn);
  const float wvn = (r < 8) ? wv[r] : 0.0f;   // pad heads 8..15 with zeros

  v8f   accA = {}, accB = {};           // WMMA D: numerators for both groups
  float zA = 0.0f, zB = 0.0f;           // softmax denominators

  for (int st = 0; st < 4; ++st) {
    __syncthreads();
    const int base = st * STILE;
    for (int t = tid; t < STILE; t += 256) {
      x1s[t] = x1[base + t]; x2s[t] = x2[base + t];
    }
    __syncthreads();
    const int lb = w * 256;             // each wave owns a 256-column sub-slab
    for (int lc = 0; lc < 256; lc += 4) {
      const int j0l = lb + lc + kk;     // local col of this lane's K=kk slot
      const float x2a = x2s[j0l], x2b = x2s[j0l + 1];
      // unnormalized softmax weights e = exp(score - rowmax), via v_exp_f32
      const float e0A = __builtin_amdgcn_exp2f(fmaf(biA, x2a, -rmA));
      const float e1A = __builtin_amdgcn_exp2f(fmaf(biA, x2b, -rmA));
      const float e0B = __builtin_amdgcn_exp2f(fmaf(biB, x2a, -rmB));
      const float e1B = __builtin_amdgcn_exp2f(fmaf(biB, x2b, -rmB));
      zA += e0A + e1A;                  // denominators sum UNmasked e
      zB += e0B + e1B;
      // one threefry call -> dropout bits for rows iA AND iB at column j
      const unsigned f0 = iA * (unsigned)NCOL + (unsigned)(base + j0l);
      const uint2 p0 = jax_threefry_pair(f0);
      const uint2 p1 = jax_threefry_pair(f0 + 1u);
      v2f Aa, Ab, B;
      Aa.x = keep_bits(p0.x) ? e0A : 0.0f;    // dropout on numerator only
      Aa.y = keep_bits(p1.x) ? e1A : 0.0f;
      Ab.x = keep_bits(p0.y) ? e0B : 0.0f;
      Ab.y = keep_bits(p1.y) ? e1B : 0.0f;
      B.x  = x1s[j0l]     * wvn;        // v[j, h] = x1[j] * wv[h] (shared)
      B.y  = x1s[j0l + 1] * wvn;
      // D[16x16] += A[16x4] x B[4x16], accumulated over all K chunks
      accA = __builtin_amdgcn_wmma_f32_16x16x4_f32(
          false, Aa, false, B, (short)0, accA, false, false);
      accB = __builtin_amdgcn_wmma_f32_16x16x4_f32(
          false, Ab, false, B, (short)0, accB, false, false);
    }
  }

  // ---- deterministic cross-wave reduction through LDS ----
  __syncthreads();
#pragma unroll
  for (int d = 0; d < 8; ++d) {
    accS[tid * 8 + d]        = accA[d];
    accS[2048 + tid * 8 + d] = accB[d];
  }
  zS[tid]       = zA;
  zS[256 + tid] = zB;
  __syncthreads();

  // 256 threads -> 32 rows x 8 heads, one output each
  {
    const int M32 = tid >> 3, h = tid & 7;
    const int set = M32 >> 4;           // 0: rows rowbase+M, 1: +4096
    const int M   = M32 & 15;
    const int srcLane = ((M >> 3) << 4) + h;   // D layout: M = d + 8*(lane/16)
    const int d = M & 7;
    const int ab = set * 2048, zb = set * 256;
    float num = 0.f, zt = 0.f;
#pragma unroll
    for (int ww = 0; ww < 8; ++ww) {
      num += accS[ab + (ww * 32 + srcLane) * 8 + d];
      zt  += zS[zb + ww * 32 + M] + zS[zb + ww * 32 + M + 16];
    }
    const unsigned row = blockIdx.x * ROWS_PER_BLOCK + M + (set ? 4096u : 0u);
    // out = (1/keep) * num / Z  ==  num / (0.9 * Z)
    out[row * 8u + h] = num / (0.9f * zt);
  }
}

extern "C" void kernel_launch(void* const* d_in, const int* in_sizes, int n_in,
                              void* d_out, int out_size, void* d_ws, size_t ws_size,
                              hipStream_t stream) {
  (void)in_sizes; (void)n_in; (void)out_size; (void)d_ws; (void)ws_size;
  const float* x1 = (const float*)d_in[0];   // [8192]  (DIM=1)
  const float* x2 = (const float*)d_in[1];   // [8192]
  const float* wq = (const float*)d_in[2];   // [8]
  const float* wk = (const float*)d_in[3];   // [8]
  const float* wv = (const float*)d_in[4];   // [8]
  float* out = (float*)d_out;                // [8192, 8]
  // 256 blocks x (16 + 16 paired) rows = 8192 rows;
  // 8 waves/block split the 8192 columns; threefry shared across row pairs.
  attn_dim1_kernel<<<256, 256, 0, stream>>>(x1, x2, wq, wk, wv, out);
}